// GNN_17592186044939
// MI455X (gfx1250) — compile-verified
//
#include <hip/hip_runtime.h>
#include <hip/hip_bf16.h>

typedef __attribute__((ext_vector_type(2))) float v2f;
typedef __attribute__((ext_vector_type(8))) float v8f;

// ---------------------------------------------------------------------------
// Degree kernels: deg[i] starts at 1 (self loop), += 1 per incoming edge,
// then dinv[i] = rsqrt(deg[i]) (deg >= 1 always, no zero guard needed).
// ---------------------------------------------------------------------------
__global__ void k_deg_init(float* __restrict__ deg, int n) {
  int i = blockIdx.x * blockDim.x + threadIdx.x;
  if (i < n) deg[i] = 1.0f;
}

__global__ void k_deg_count(const int* __restrict__ dst, int E, float* __restrict__ deg) {
  int e = blockIdx.x * blockDim.x + threadIdx.x;
  if (e < E) atomicAdd(&deg[dst[e]], 1.0f);
}

__global__ void k_deg_rsqrt(float* __restrict__ deg, int n) {
  int i = blockIdx.x * blockDim.x + threadIdx.x;
  if (i < n) deg[i] = rsqrtf(deg[i]);
}

// ---------------------------------------------------------------------------
// Fused GEMM + dinv pre-scale using V_WMMA_F32_16X16X4_F32 (wave32).
//   hs[r, c]  = (X @ W)[r, c] * dinv[r]        (gather source)
//   agg[r, c] = hs[r, c]                        (self-loop init of accumulator)
// One wave per 16-row tile; NT 16-wide column tiles (N = NT*16); K stepped by 4.
// A 16x4 f32 frag: lanes 0-15 -> K = k0,k0+1 ; lanes 16-31 -> K = k0+2,k0+3.
// W is staged in LDS pre-paired along K: Wl2[p*N + n] = {W[2p][n], W[2p+1][n]}
// so each lane's B fragment is ONE aligned ds_load_b64 into the even VGPR
// pair the WMMA consumes (no repacking moves in the hot loop).
// nrows is a multiple of 16 here (100000 = 6250*16): no tails, EXEC all-1s.
// ---------------------------------------------------------------------------
template<int K, int NT>
__global__ void k_gemm_wmma(const float* __restrict__ X, const float* __restrict__ W,
                            const float* __restrict__ dinv,
                            float* __restrict__ hs, float* __restrict__ agg, int nrows) {
  constexpr int N = NT * 16;
  __shared__ v2f Wl2[(K / 2) * N];
  for (int i = threadIdx.x; i < (K / 2) * N; i += blockDim.x) {
    const int p  = i / N;          // K-pair index
    const int nn = i - p * N;      // column
    v2f w;
    w.x = W[(2 * p)     * N + nn];
    w.y = W[(2 * p + 1) * N + nn];
    Wl2[i] = w;
  }
  __syncthreads();

  const int wave = threadIdx.x >> 5;
  const int lane = threadIdx.x & 31;
  const int tile = blockIdx.x * (blockDim.x >> 5) + wave;
  const int r0 = tile * 16;
  if (r0 >= nrows) return;  // uniform per wave (after the barrier)

  const int halfsel = lane >> 4;        // 0: lanes 0-15, 1: lanes 16-31
  const int koff    = halfsel * 2;      // K offset within the 4-wide step
  const int mcol    = lane & 15;        // M index for A, N index for B/C
  const float* xrow = X + (long long)(r0 + mcol) * K;

  v8f c[NT] = {};
#pragma unroll
  for (int k0 = 0; k0 < K; k0 += 4) {
    v2f a = *(const v2f*)(xrow + k0 + koff);            // one global_load_b64
    const int prow = (k0 >> 1) + halfsel;               // K-pair row in LDS
#pragma unroll
    for (int t = 0; t < NT; ++t) {
      v2f b = Wl2[prow * N + t * 16 + mcol];            // one ds_load_b64
      c[t] = __builtin_amdgcn_wmma_f32_16x16x4_f32(
          /*neg_a=*/false, a, /*neg_b=*/false, b,
          /*c_mod=*/(short)0, c[t], /*reuse_a=*/false, /*reuse_b=*/false);
    }
  }

  const int rowBase = r0 + halfsel * 8;
#pragma unroll
  for (int j = 0; j < 8; ++j) {
    const int row = rowBase + j;
    const float dv = dinv[row];
#pragma unroll
    for (int t = 0; t < NT; ++t) {
      const float v = c[t][j] * dv;
      const long long o = (long long)row * N + t * 16 + mcol;
      hs[o]  = v;
      agg[o] = v;
    }
  }
}

// ---------------------------------------------------------------------------
// Edge scatter: one thread per (edge, feature). Consecutive lanes share an
// edge and walk the feature row -> coalesced gather of hs[s,:] (L2-resident)
// and contiguous f32 atomics into agg[d,:]. F = 1<<LOGF (32 or 16).
// ---------------------------------------------------------------------------
template<int LOGF>
__global__ void k_scatter(const int* __restrict__ src, const int* __restrict__ dst,
                          const float* __restrict__ hs, float* __restrict__ agg,
                          long long total) {
  const long long idx = (long long)blockIdx.x * blockDim.x + threadIdx.x;
  if (idx >= total) return;
  const int e = (int)(idx >> LOGF);
  const int j = (int)(idx & ((1 << LOGF) - 1));
  const int s = src[e];
  const int d = dst[e];
  atomicAdd(agg + (((long long)d) << LOGF) + j, hs[(((long long)s) << LOGF) + j]);
}

// ---------------------------------------------------------------------------
// Epilogues (in place): layer1 -> relu(dinv*agg + b), layer2 -> dinv*agg + b.
// ---------------------------------------------------------------------------
template<int LOGF, bool RELU>
__global__ void k_epilogue(float* __restrict__ agg, const float* __restrict__ dinv,
                           const float* __restrict__ bias, long long total) {
  const long long idx = (long long)blockIdx.x * blockDim.x + threadIdx.x;
  if (idx >= total) return;
  const int i = (int)(idx >> LOGF);
  const int j = (int)(idx & ((1 << LOGF) - 1));
  float v = fmaf(dinv[i], agg[idx], bias[j]);
  agg[idx] = RELU ? fmaxf(v, 0.0f) : v;
}

// ---------------------------------------------------------------------------
// Orchestration. Inputs: x[N,128] f32, edge_index[2,E] i32, W1[128,32], b1[32],
// W2[32,16], b2[16]. Output: [N,16] f32. d_out doubles as the layer-2
// accumulator. agg1 is reused in place as the layer-2 input after the ReLU.
// ---------------------------------------------------------------------------
extern "C" void kernel_launch(void* const* d_in, const int* in_sizes, int n_in,
                              void* d_out, int out_size, void* d_ws, size_t ws_size,
                              hipStream_t stream) {
  const float* x  = (const float*)d_in[0];
  const int*   ei = (const int*)d_in[1];
  const float* W1 = (const float*)d_in[2];
  const float* b1 = (const float*)d_in[3];
  const float* W2 = (const float*)d_in[4];
  const float* b2 = (const float*)d_in[5];
  float* out = (float*)d_out;

  const int n = in_sizes[0] / 128;   // 100000
  const int E = in_sizes[1] / 2;     // 3200000
  const int* src = ei;
  const int* dst = ei + E;

  float* ws   = (float*)d_ws;
  float* dinv = ws;                                  // n
  float* hs1  = dinv + n;                            // n*32
  float* agg1 = hs1 + (size_t)n * 32;                // n*32  (becomes layer-2 input)
  float* hs2  = agg1 + (size_t)n * 32;               // n*16

  const int TB = 256;  // 8 waves/block on wave32

  // Degrees -> dinv
  k_deg_init <<<(n + TB - 1) / TB, TB, 0, stream>>>(dinv, n);
  k_deg_count<<<(E + TB - 1) / TB, TB, 0, stream>>>(dst, E, dinv);
  k_deg_rsqrt<<<(n + TB - 1) / TB, TB, 0, stream>>>(dinv, n);

  const int tiles = (n + 15) / 16;
  const int wpb   = TB / 32;
  const int gblk  = (tiles + wpb - 1) / wpb;

  // Layer 1
  k_gemm_wmma<128, 2><<<gblk, TB, 0, stream>>>(x, W1, dinv, hs1, agg1, n);
  {
    long long total = (long long)E * 32;
    k_scatter<5><<<(unsigned)((total + TB - 1) / TB), TB, 0, stream>>>(src, dst, hs1, agg1, total);
  }
  {
    long long total = (long long)n * 32;
    k_epilogue<5, true><<<(unsigned)((total + TB - 1) / TB), TB, 0, stream>>>(agg1, dinv, b1, total);
  }

  // Layer 2 (reads agg1 as [n,32] input; accumulates directly in d_out)
  k_gemm_wmma<32, 1><<<gblk, TB, 0, stream>>>(agg1, W2, dinv, hs2, out, n);
  {
    long long total = (long long)E * 16;
    k_scatter<4><<<(unsigned)((total + TB - 1) / TB), TB, 0, stream>>>(src, dst, hs2, out, total);
  }
  {
    long long total = (long long)n * 16;
    k_epilogue<4, false><<<(unsigned)((total + TB - 1) / TB), TB, 0, stream>>>(out, dinv, b2, total);
  }
}